// FloorPlanLoss_64450279244066
// MI455X (gfx1250) — compile-verified
//
#include <hip/hip_runtime.h>
#include <hip/hip_bf16.h>

typedef float v2f __attribute__((ext_vector_type(2)));
typedef float v8f __attribute__((ext_vector_type(8)));
typedef unsigned int v4u __attribute__((ext_vector_type(4)));
typedef int v4i __attribute__((ext_vector_type(4)));
typedef int v8i __attribute__((ext_vector_type(8)));

#define RR 192
#define NBATCH 1024
#define MSE_BLOCKS 256
#define N_ELEMS (NBATCH * RR * 2) /* 393216, divisible by 64 */
#define NBITWORDS ((RR * RR) / 32) /* 1152 */

// ws layout (as float/uint32 slots):
//   [0,    256)  : position sq-diff partials (one per MSE block)
//   [256,  512)  : size sq-diff partials
//   [512,  1536) : overlap partials (one per batch block)
//   [1536, 2560) : adjacency partials (one per batch block)
//   [2560, 3712) : triu&adj bitmask (1152 uint32 words, 1 bit per (i,j))

#if defined(__gfx1250__) && __has_builtin(__builtin_amdgcn_tensor_load_to_lds)
#define HAVE_TDM 1
// Issue a 1-D Tensor-Data-Mover copy of `ndw` dwords (global -> LDS).
// D# bitfields per CDNA5 ISA ch.8 (count=1, data_size=4B, type=2).
__device__ __forceinline__ void tdm_copy_dwords(unsigned lds_addr,
                                                unsigned long long gaddr,
                                                unsigned ndw) {
  v4u g0;
  g0.x = 1u;                                   // count=1 (valid descriptor)
  g0.y = lds_addr;                             // lds_addr
  g0.z = (unsigned)(gaddr & 0xFFFFFFFFu);      // global_addr[31:0]
  g0.w = (unsigned)((gaddr >> 32) & 0x01FFFFFFu) | (2u << 30); // [56:32]|type=2
  v8i g1;
  g1[0] = 0x00020000;                          // mask=0, data_size=2 (4 bytes)
  g1[1] = (int)(ndw << 16);                    // tensor_dim0[15:0]
  g1[2] = (int)(((ndw >> 16) & 0xFFFFu) | (1u << 16)); // dim0 hi | tensor_dim1=1
  g1[3] = (int)(ndw << 16);                    // tensor_dim1 hi=0 | tile_dim0
  g1[4] = 1;                                   // tile_dim1=1, tile_dim2=0
  g1[5] = (int)ndw;                            // tensor_dim0_stride lo
  g1[6] = (int)(ndw << 16);                    // stride0 hi=0 | stride1 lo
  g1[7] = 0;
  v4i z4 = {};
#if __clang_major__ >= 23
  v8i z8 = {};
  __builtin_amdgcn_tensor_load_to_lds(g0, g1, z4, z4, z8, 0);
#else
  __builtin_amdgcn_tensor_load_to_lds(g0, g1, z4, z4, 0);
#endif
}
#endif

// ---------------------------------------------------------------------------
// Kernel 0: compress triu & (adj>0) into a bitmask (1 bit per pair).
// ---------------------------------------------------------------------------
__global__ void fp_adj_kernel(const int* __restrict__ adj,
                              unsigned* __restrict__ bits) {
#if defined(__gfx1250__)
  int w = blockIdx.x * blockDim.x + threadIdx.x;
  if (w < NBITWORDS) {
    unsigned m = 0;
    int k0 = w * 32;
    for (int t = 0; t < 32; ++t) {
      int k = k0 + t;
      int i = k / RR;
      int j = k - i * RR;
      if (j > i && adj[k] > 0) m |= (1u << t);
    }
    bits[w] = m;
  }
#endif
}

// ---------------------------------------------------------------------------
// Kernel 1: MSE sums via WMMA-f32 reduction.
// Each wave eats 64 floats per array per iteration (float2 per lane); squared
// diffs go into the 16x4 f32 A-matrix, accumulated with B = ones into a 16x16
// f32 C tile (D[m,n] = row-sums, identical for every column n).
// Wave total = (sum of 8 D-VGPRs in lane n) + (same in lane n+16) -> xor-16.
// ---------------------------------------------------------------------------
__global__ void fp_mse_kernel(const float* __restrict__ pos,
                              const float* __restrict__ siz,
                              const float* __restrict__ tpos,
                              const float* __restrict__ tsiz,
                              float* __restrict__ ws) {
#if defined(__gfx1250__)
  const int lane = threadIdx.x & 31;
  const int wave = threadIdx.x >> 5;
  const int wavesPerBlock = blockDim.x >> 5;
  const int gwave = blockIdx.x * wavesPerBlock + wave;
  const int nwaves = gridDim.x * wavesPerBlock;
  const int N2 = N_ELEMS / 2;

  const float2* p2 = (const float2*)pos;
  const float2* s2 = (const float2*)siz;
  const float2* q2 = (const float2*)tpos;
  const float2* t2 = (const float2*)tsiz;

  float tp, ts;
#if __has_builtin(__builtin_amdgcn_wmma_f32_16x16x4_f32)
  v8f cp = {}; v8f cs = {};
  v2f ones; ones.x = 1.0f; ones.y = 1.0f;
  for (int base = gwave * 32; base < N2; base += nwaves * 32) {
    int i0 = base + lane;
    float2 a = p2[i0], b = q2[i0];
    float2 c = s2[i0], d = t2[i0];
    float dpx = a.x - b.x, dpy = a.y - b.y;
    float dsx = c.x - d.x, dsy = c.y - d.y;
    v2f ap; ap.x = dpx * dpx; ap.y = dpy * dpy;
    v2f as; as.x = dsx * dsx; as.y = dsy * dsy;
    cp = __builtin_amdgcn_wmma_f32_16x16x4_f32(false, ap, false, ones,
                                               (short)0, cp, false, false);
    cs = __builtin_amdgcn_wmma_f32_16x16x4_f32(false, as, false, ones,
                                               (short)0, cs, false, false);
  }
  tp = cp[0] + cp[1] + cp[2] + cp[3] + cp[4] + cp[5] + cp[6] + cp[7];
  ts = cs[0] + cs[1] + cs[2] + cs[3] + cs[4] + cs[5] + cs[6] + cs[7];
  tp += __shfl_xor(tp, 16, 32);
  ts += __shfl_xor(ts, 16, 32);
#else
  tp = 0.0f; ts = 0.0f;
  for (int base = gwave * 32; base < N2; base += nwaves * 32) {
    int i0 = base + lane;
    float2 a = p2[i0], b = q2[i0];
    float2 c = s2[i0], d = t2[i0];
    float dpx = a.x - b.x, dpy = a.y - b.y;
    float dsx = c.x - d.x, dsy = c.y - d.y;
    tp += dpx * dpx + dpy * dpy;
    ts += dsx * dsx + dsy * dsy;
  }
  for (int m = 16; m > 0; m >>= 1) {
    tp += __shfl_xor(tp, m, 32);
    ts += __shfl_xor(ts, m, 32);
  }
#endif

  __shared__ float red[2][8];
  if (lane == 0) { red[0][wave] = tp; red[1][wave] = ts; }
  __syncthreads();
  if (threadIdx.x == 0) {
    float sp = 0.0f, ss = 0.0f;
    for (int i = 0; i < wavesPerBlock; ++i) { sp += red[0][i]; ss += red[1][i]; }
    ws[blockIdx.x] = sp;
    ws[MSE_BLOCKS + blockIdx.x] = ss;
  }
#endif
}

// ---------------------------------------------------------------------------
// Kernel 2: one block per batch. TDM-stage this batch's pos/size rows
// (1536 B each) and the adjacency bitmask (4608 B) into LDS, then sweep all
// i<j pairs entirely out of LDS.
// ---------------------------------------------------------------------------
__global__ void fp_pair_kernel(const float* __restrict__ pos,
                               const float* __restrict__ siz,
                               const unsigned* __restrict__ bits,
                               float* __restrict__ ws) {
#if defined(__gfx1250__)
  __shared__ float sh_pos[2 * RR];        // x,y interleaved
  __shared__ float sh_siz[2 * RR];        // w,h interleaved
  __shared__ unsigned sh_bits[NBITWORDS]; // triu & adj bitmask
  __shared__ float red[2][8];

  const int b = blockIdx.x;
  const int rowBase = b * (2 * RR);

#ifdef HAVE_TDM
  if (threadIdx.x < 32) {
    tdm_copy_dwords((unsigned)(unsigned long long)(void*)sh_pos,
                    (unsigned long long)(const void*)(pos + rowBase), 2 * RR);
    tdm_copy_dwords((unsigned)(unsigned long long)(void*)sh_siz,
                    (unsigned long long)(const void*)(siz + rowBase), 2 * RR);
    tdm_copy_dwords((unsigned)(unsigned long long)(void*)sh_bits,
                    (unsigned long long)(const void*)bits, NBITWORDS);
    __builtin_amdgcn_s_wait_tensorcnt(0);
  }
#else
  for (int t = threadIdx.x; t < 2 * RR; t += blockDim.x) {
    sh_pos[t] = pos[rowBase + t];
    sh_siz[t] = siz[rowBase + t];
  }
  for (int t = threadIdx.x; t < NBITWORDS; t += blockDim.x)
    sh_bits[t] = bits[t];
#endif
  __syncthreads();

  float ov = 0.0f, ad = 0.0f;
  for (int k = threadIdx.x; k < RR * RR; k += blockDim.x) {
    int i = k / RR;
    int j = k - i * RR;
    if (j > i) {
      float xi = sh_pos[2 * i], yi = sh_pos[2 * i + 1];
      float wi = sh_siz[2 * i], hi = sh_siz[2 * i + 1];
      float xj = sh_pos[2 * j], yj = sh_pos[2 * j + 1];
      float wj = sh_siz[2 * j], hj = sh_siz[2 * j + 1];
      float ow = fminf(xi + wi, xj + wj) - fmaxf(xi, xj);
      float oh = fminf(yi + hi, yj + hj) - fmaxf(yi, yj);
      ow = fmaxf(ow, 0.0f);
      oh = fmaxf(oh, 0.0f);
      ov += ow * oh;
      if ((sh_bits[k >> 5] >> (k & 31)) & 1u) {
        float dx = (xi + 0.5f * wi) - (xj + 0.5f * wj);
        float dy = (yi + 0.5f * hi) - (yj + 0.5f * hj);
        ad += sqrtf(dx * dx + dy * dy);
      }
    }
  }
  for (int m = 16; m > 0; m >>= 1) {
    ov += __shfl_xor(ov, m, 32);
    ad += __shfl_xor(ad, m, 32);
  }
  const int lane = threadIdx.x & 31;
  const int wave = threadIdx.x >> 5;
  if (lane == 0) { red[0][wave] = ov; red[1][wave] = ad; }
  __syncthreads();
  if (threadIdx.x == 0) {
    float so = 0.0f, sa = 0.0f;
    for (int i = 0; i < (int)(blockDim.x >> 5); ++i) { so += red[0][i]; sa += red[1][i]; }
    ws[512 + b] = so;
    ws[1536 + b] = sa;
  }
#endif
}

// ---------------------------------------------------------------------------
// Kernel 3: deterministic finalize (single block).
// ---------------------------------------------------------------------------
__global__ void fp_final_kernel(const float* __restrict__ ws,
                                float* __restrict__ out) {
#if defined(__gfx1250__)
  __shared__ float acc[4][256];
  const int t = threadIdx.x;
  float sp = ws[t];
  float ss = ws[256 + t];
  float so = 0.0f, sa = 0.0f;
  for (int k = t; k < NBATCH; k += 256) {
    so += ws[512 + k];
    sa += ws[1536 + k];
  }
  acc[0][t] = sp; acc[1][t] = ss; acc[2][t] = so; acc[3][t] = sa;
  __syncthreads();
  for (int s = 128; s > 0; s >>= 1) {
    if (t < s)
      for (int q = 0; q < 4; ++q) acc[q][t] += acc[q][t + s];
    __syncthreads();
  }
  if (t == 0) {
    const float invN = 1.0f / (float)N_ELEMS;
    const float invB = 1.0f / (float)NBATCH;
    float pos_loss  = acc[0][0] * invN;
    float size_loss = acc[1][0] * invN;
    float overlap   = acc[2][0] * invB;
    float adjl      = acc[3][0] * invB;
    out[0] = pos_loss + size_loss + 0.5f * overlap + 0.3f * adjl;
    out[1] = pos_loss;
    out[2] = size_loss;
    out[3] = overlap;
    out[4] = adjl;
  }
#endif
}

extern "C" void kernel_launch(void* const* d_in, const int* in_sizes, int n_in,
                              void* d_out, int out_size, void* d_ws, size_t ws_size,
                              hipStream_t stream) {
  const float* pos  = (const float*)d_in[0];
  const float* siz  = (const float*)d_in[1];
  const float* tpos = (const float*)d_in[2];
  const float* tsiz = (const float*)d_in[3];
  const int*   adj  = (const int*)d_in[4];
  float* ws = (float*)d_ws;
  unsigned* bits = (unsigned*)ws + 2560;
  float* out = (float*)d_out;

  fp_adj_kernel<<<(NBITWORDS + 255) / 256, 256, 0, stream>>>(adj, bits);
  fp_mse_kernel<<<MSE_BLOCKS, 256, 0, stream>>>(pos, siz, tpos, tsiz, ws);
  fp_pair_kernel<<<NBATCH, 256, 0, stream>>>(pos, siz, bits, ws);
  fp_final_kernel<<<1, 256, 0, stream>>>(ws, out);
}